// CrostonsRNN_42631845380193
// MI455X (gfx1250) — compile-verified
//
#include <hip/hip_runtime.h>
#include <cstdint>

// ---------------- problem geometry (matches reference) ----------------
#define TSTEPS 2048              // T
#define LANES 32                 // wave32: chunks per row
#define CHUNK 64                 // time steps per lane (32*64 = 2048)
#define ROWS_PER_BLOCK 8         // one wave per row
#define THREADS (LANES * ROWS_PER_BLOCK)
#define CHUNK_DW (CHUNK + 2)     // 66-dword stride: bank = (2i+t)%64, conflict-free; 8B aligned

// payload type for the b64 async ops: 64-bit vector element, per clang diagnostic
typedef int v2i __attribute__((vector_size(2 * sizeof(int))));
typedef __attribute__((address_space(1))) v2i g_v2i;   // global (AS1)
typedef __attribute__((address_space(3))) v2i l_v2i;   // LDS (AS3)
typedef __attribute__((address_space(3))) void las_void;

// ---------------- CDNA5 async global<->LDS helpers ----------------
__device__ __forceinline__ void async_load_b64(const float* g, float* l) {
#if __has_builtin(__builtin_amdgcn_global_load_async_to_lds_b64)
  __builtin_amdgcn_global_load_async_to_lds_b64((g_v2i*)g, (l_v2i*)l, 0, 0);
#else
  unsigned la = (unsigned)(uintptr_t)(las_void*)l;
  asm volatile("global_load_async_to_lds_b64 %0, %1, off"
               :: "v"(la), "v"(g) : "memory");
#endif
}

__device__ __forceinline__ void async_store_b64(float* g, float* l) {
#if __has_builtin(__builtin_amdgcn_global_store_async_from_lds_b64)
  __builtin_amdgcn_global_store_async_from_lds_b64((g_v2i*)g, (l_v2i*)l, 0, 0);
#else
  unsigned la = (unsigned)(uintptr_t)(las_void*)l;
  asm volatile("global_store_async_from_lds_b64 %0, %1, off"
               :: "v"(g), "v"(la) : "memory");
#endif
}

__device__ __forceinline__ void wait_async0() {
#if __has_builtin(__builtin_amdgcn_s_wait_asynccnt)
  __builtin_amdgcn_s_wait_asynccnt(0);
#else
  asm volatile("s_wait_asynccnt 0" ::: "memory");
#endif
  asm volatile("" ::: "memory");   // compiler ordering fence for subsequent LDS reads
}

__device__ __forceinline__ void wait_ds0() {
#if __has_builtin(__builtin_amdgcn_s_wait_dscnt)
  __builtin_amdgcn_s_wait_dscnt(0);
#else
  asm volatile("s_wait_dscnt 0" ::: "memory");
#endif
  asm volatile("" ::: "memory");
}

// wave32 shuffle-up via ds_bpermute (index in bytes; bits[6:2] used)
__device__ __forceinline__ float shflu(float v, int delta, int lane) {
  int src = lane - delta;
  src = (src < 0) ? lane : src;    // clamped; result unused when lane < delta
  return __int_as_float(__builtin_amdgcn_ds_bpermute(src << 2, __float_as_int(v)));
}

// ---------------- kernel ----------------
__global__ __launch_bounds__(THREADS)
void croston_scan_kernel(const float* __restrict__ x,
                         const float* __restrict__ alpha,
                         const float* __restrict__ Z0,
                         const float* __restrict__ V0,
                         const float* __restrict__ q0,
                         float* __restrict__ out, int nrows) {
  __shared__ float tile[ROWS_PER_BLOCK * LANES * CHUNK_DW];  // ~66 KB

  const int tid  = threadIdx.x;
  const int lane = tid & (LANES - 1);
  const int wave = tid >> 5;
  const int row  = blockIdx.x * ROWS_PER_BLOCK + wave;
  if (row >= nrows) return;

  const float a = alpha[0];        // uniform -> scalar load
  const float u = 1.0f - a;

  float* lds = &tile[(wave * LANES + lane) * CHUNK_DW];
  const float* gsrc = x   + (size_t)row * TSTEPS + lane * CHUNK;
  float*       gdst = out + (size_t)row * TSTEPS + lane * CHUNK;

  // ---- stage this wave's 8KB row into LDS via async DMA (ASYNCcnt) ----
#pragma unroll
  for (int j = 0; j < CHUNK / 2; ++j)
    async_load_b64(gsrc + 2 * j, lds + 2 * j);
  wait_async0();

  // ---- phase 1: compose per-chunk affine transform ----
  // Z'      = cz*Z + dz
  // [V,q]'  = A*[V,q] + b   (A = [[a00,a01],[a10,a11]], b = [b0,b1])
  float cz = 1.f, dz = 0.f;
  float a00 = 1.f, a01 = 0.f, a10 = 0.f, a11 = 1.f, b0 = 0.f, b1 = 0.f;
#pragma unroll 4
  for (int t = 0; t < CHUNK; ++t) {
    const float xt = lds[t];
    const bool nz = (xt != 0.0f);
    // step: nz -> A_t=[[u,a],[0,0]], b_t=[0,1], cz_t=u, dz_t=a*xt
    //       z  -> identity A, b_t=[0,1]
    const float ndz = u * dz + a * xt;
    const float ncz = u * cz;
    const float n00 = u * a00 + a * a10;
    const float n01 = u * a01 + a * a11;
    const float nb0 = u * b0 + a * b1;
    cz  = nz ? ncz : cz;
    dz  = nz ? ndz : dz;
    a00 = nz ? n00 : a00;
    a01 = nz ? n01 : a01;
    a10 = nz ? 0.f : a10;
    a11 = nz ? 0.f : a11;
    b0  = nz ? nb0 : b0;
    b1  = nz ? 1.f : (b1 + 1.f);
  }

  // ---- phase 1.5: wave-inclusive scan of transforms (composition) ----
#pragma unroll
  for (int d = 1; d < LANES; d <<= 1) {
    const float pcz = shflu(cz, d, lane),  pdz = shflu(dz, d, lane);
    const float p00 = shflu(a00, d, lane), p01 = shflu(a01, d, lane);
    const float p10 = shflu(a10, d, lane), p11 = shflu(a11, d, lane);
    const float pb0 = shflu(b0, d, lane),  pb1 = shflu(b1, d, lane);
    if (lane >= d) {  // this = this ∘ prev  (prev applied first)
      dz = cz * pdz + dz;
      cz = cz * pcz;
      const float t00 = a00 * p00 + a01 * p10;
      const float t01 = a00 * p01 + a01 * p11;
      const float t10 = a10 * p00 + a11 * p10;
      const float t11 = a10 * p01 + a11 * p11;
      b0 = a00 * pb0 + a01 * pb1 + b0;
      b1 = a10 * pb0 + a11 * pb1 + b1;
      a00 = t00; a01 = t01; a10 = t10; a11 = t11;
    }
  }

  // ---- exclusive state at chunk start: apply lane-1 inclusive to init ----
  const float ecz = shflu(cz, 1, lane),  edz = shflu(dz, 1, lane);
  const float e00 = shflu(a00, 1, lane), e01 = shflu(a01, 1, lane);
  const float e10 = shflu(a10, 1, lane), e11 = shflu(a11, 1, lane);
  const float eb0 = shflu(b0, 1, lane),  eb1 = shflu(b1, 1, lane);
  const float Zi = Z0[row], Vi = V0[row], qi = q0[row];
  float Z, V, q;
  if (lane == 0) {
    Z = Zi; V = Vi; q = qi;
  } else {
    Z = ecz * Zi + edz;
    V = e00 * Vi + e01 * qi + eb0;
    q = e10 * Vi + e11 * qi + eb1;
  }

  // ---- phase 2: exact reference recurrence within chunk; out -> LDS ----
#pragma unroll 4
  for (int t = 0; t < CHUNK; ++t) {
    const float xt = lds[t];
    const bool nz = (xt != 0.0f);
    const float Zn = a * xt + u * Z;
    const float Vn = a * q + u * V;
    Z = nz ? Zn : Z;
    V = nz ? Vn : V;
    q = nz ? 1.f : (q + 1.f);
    lds[t] = Z / V;
  }
  wait_ds0();  // LDS writes visible before async drain

  // ---- drain outputs LDS -> global via async DMA; s_endpgm waits counters ----
#pragma unroll
  for (int j = 0; j < CHUNK / 2; ++j)
    async_store_b64(gdst + 2 * j, lds + 2 * j);
}

// ---------------- launch ----------------
extern "C" void kernel_launch(void* const* d_in, const int* in_sizes, int n_in,
                              void* d_out, int out_size, void* d_ws, size_t ws_size,
                              hipStream_t stream) {
  const float* x     = (const float*)d_in[0];
  const float* alpha = (const float*)d_in[1];
  const float* Z0    = (const float*)d_in[2];
  const float* V0    = (const float*)d_in[3];
  const float* q0    = (const float*)d_in[4];
  float* out = (float*)d_out;

  const int nrows = in_sizes[2];  // Z0 has B elements
  const int blocks = (nrows + ROWS_PER_BLOCK - 1) / ROWS_PER_BLOCK;
  croston_scan_kernel<<<blocks, THREADS, 0, stream>>>(x, alpha, Z0, V0, q0, out, nrows);
}